// GraphUnet_4861902979240
// MI455X (gfx1250) — compile-verified
//
#include <hip/hip_runtime.h>
#include <math.h>

// ---------------------------------------------------------------------------
// Graph U-Net for MI455X (gfx1250).  fp32 WMMA (V_WMMA_F32_16X16X4_F32) for
// the projection GEMMs (xp = X@W+b) and the attention GEMMs (att@xp).
// Branch-free, software-pipelined WMMA mainloop with strength-reduced
// pointers: loads for k-step i+1 are issued before the WMMAs of k-step i.
// Each wave computes a 16x32 stripe (2 accumulators) for occupancy.
// ---------------------------------------------------------------------------

typedef float v2f __attribute__((ext_vector_type(2)));
typedef float v8f __attribute__((ext_vector_type(8)));

#define N0 4096
#define N1 3276   // int(0.8*4096)
#define N2 1965   // int(0.6*3276)
#define FDIM 320

static __device__ __forceinline__ float warp_sum(float v) {
#pragma unroll
  for (int off = 16; off > 0; off >>= 1) v += __shfl_xor(v, off, 32);
  return v;
}
static __device__ __forceinline__ float warp_max(float v) {
#pragma unroll
  for (int off = 16; off > 0; off >>= 1) v = fmaxf(v, __shfl_xor(v, off, 32));
  return v;
}
static __device__ __forceinline__ int warp_sum_i(int v) {
#pragma unroll
  for (int off = 16; off > 0; off >>= 1) v += __shfl_xor(v, off, 32);
  return v;
}

static __device__ __forceinline__ v8f wmma4(v2f a, v2f b, v8f c) {
  return __builtin_amdgcn_wmma_f32_16x16x4_f32(
      false, a, false, b, (short)0, c, false, false);
}

// --------- branch-free, pipelined fp32 WMMA 16x32 stripe mainloop ----------
// acc[2] covers cols [n0, n0+32).  A rows may be garbage beyond M (allocation
// padded; garbage stays confined to its own output row); K and lda must be
// multiples of 4; B must have K valid rows.
static __device__ __forceinline__ void wmma_stripe_f32(
    const float* __restrict__ A, int lda,
    const float* __restrict__ B, int ldb,
    int K, int m0, int n0, int lane, v8f acc[2]) {
  const int lo = lane & 15;
  const int hi = lane >> 4;
  const float* Ap = A + (size_t)(m0 + lo) * lda + 2 * hi;   // A[m][ka..ka+1]
  const float* Bp = B + (size_t)(2 * hi) * ldb + n0 + lo;   // B[ka][col]
  const float* Bq = Bp + ldb;                               // B[ka+1][col]
  const size_t bstep = (size_t)4 * ldb;
  acc[0] = (v8f){0.f,0.f,0.f,0.f,0.f,0.f,0.f,0.f};
  acc[1] = (v8f){0.f,0.f,0.f,0.f,0.f,0.f,0.f,0.f};

  // prologue: fragments for k-step 0
  v2f af = *(const v2f*)Ap;
  v2f b0, b1;
  b0.x = Bp[0];  b0.y = Bq[0];
  b1.x = Bp[16]; b1.y = Bq[16];

#pragma unroll 2
  for (int k0 = 4; k0 < K; k0 += 4) {
    Ap += 4; Bp += bstep; Bq += bstep;
    v2f afn = *(const v2f*)Ap;        // next k-step loads issued first
    v2f c0, c1;
    c0.x = Bp[0];  c0.y = Bq[0];
    c1.x = Bp[16]; c1.y = Bq[16];
    acc[0] = wmma4(af, b0, acc[0]);   // consume previous k-step
    acc[1] = wmma4(af, b1, acc[1]);
    af = afn; b0 = c0; b1 = c1;
  }
  acc[0] = wmma4(af, b0, acc[0]);
  acc[1] = wmma4(af, b1, acc[1]);
}

// out[MxN] = A[MxK] @ B[KxN] + bias[N]   (N % 32 == 0, K % 4 == 0)
__global__ void gemm_bias_kernel(const float* __restrict__ A,
                                 const float* __restrict__ B,
                                 const float* __restrict__ bias,
                                 float* __restrict__ out,
                                 int M, int K, int N) {
  const int wave = (blockIdx.x * blockDim.x + threadIdx.x) >> 5;
  const int lane = threadIdx.x & 31;
  const int stripes = N >> 5;
  const int tilesM = (M + 15) >> 4;
  if (wave >= tilesM * stripes) return;
  const int m0 = (wave / stripes) * 16;
  const int n0 = (wave % stripes) * 32;
  v8f acc[2];
  wmma_stripe_f32(A, K, B, N, K, m0, n0, lane, acc);
  const int lo = lane & 15, hi = lane >> 4;
#pragma unroll
  for (int t = 0; t < 2; ++t) {
    const int col = n0 + t * 16 + lo;
    const float bv = bias[col];
#pragma unroll
    for (int v = 0; v < 8; ++v) {
      const int r = m0 + v + 8 * hi;
      if (r < M) out[(size_t)r * N + col] = acc[t][v] + bv;
    }
  }
}

// out = epilogue( P[MxK,lda] @ xp[KxN] ):  /rowsum, leaky(0.01), +res, +extra
__global__ void gemm_attn_kernel(const float* __restrict__ P, int lda,
                                 const float* __restrict__ Xp,
                                 const float* __restrict__ lsum,
                                 const float* __restrict__ res,
                                 const float* __restrict__ extra,
                                 float* __restrict__ out,
                                 int M, int K, int N) {
  const int wave = (blockIdx.x * blockDim.x + threadIdx.x) >> 5;
  const int lane = threadIdx.x & 31;
  const int stripes = N >> 5;
  const int tilesM = (M + 15) >> 4;
  if (wave >= tilesM * stripes) return;
  const int m0 = (wave / stripes) * 16;
  const int n0 = (wave % stripes) * 32;
  v8f acc[2];
  wmma_stripe_f32(P, lda, Xp, N, K, m0, n0, lane, acc);
  const int lo = lane & 15, hi = lane >> 4;
#pragma unroll
  for (int t = 0; t < 2; ++t) {
    const int col = n0 + t * 16 + lo;
#pragma unroll
    for (int v = 0; v < 8; ++v) {
      const int r = m0 + v + 8 * hi;
      if (r < M) {
        float val = acc[t][v] / lsum[r];
        val = (val >= 0.f) ? val : 0.01f * val;        // F.leaky_relu default
        const size_t o = (size_t)r * N + col;
        if (res)   val += res[o];
        if (extra) val += extra[o];
        out[o] = val;
      }
    }
  }
}

// a[i] = xp[i].phi[:320] ; b[i] = xp[i].phi[320:]
__global__ void rowdot2_kernel(const float* __restrict__ xp,
                               const float* __restrict__ phi,
                               float* __restrict__ aout,
                               float* __restrict__ bout, int n) {
  const int i = (blockIdx.x * blockDim.x + threadIdx.x) >> 5;
  const int lane = threadIdx.x & 31;
  if (i >= n) return;
  const float* row = xp + (size_t)i * FDIM;
  float s1 = 0.f, s2 = 0.f;
  for (int c = lane; c < FDIM; c += 32) {
    const float v = row[c];
    s1 += v * phi[c];
    s2 += v * phi[FDIM + c];
  }
  s1 = warp_sum(s1);
  s2 = warp_sum(s2);
  if (lane == 0) { aout[i] = s1; bout[i] = s2; }
}

// row max of masked leaky(0.2) scores; diagonal always unmasked
__global__ void attn_rowmax_kernel(const float* __restrict__ Amat,
                                   const float* __restrict__ av,
                                   const float* __restrict__ bv,
                                   float* __restrict__ mout, int n) {
  const int i = (blockIdx.x * blockDim.x + threadIdx.x) >> 5;
  const int lane = threadIdx.x & 31;
  if (i >= n) return;
  const float* Ar = Amat + (size_t)i * n;
  const float ai = av[i];
  float m = -3.0e38f;
  for (int j = lane; j < n; j += 32) {
    if ((Ar[j] > 0.f) || (j == i)) {
      float s = ai + bv[j];
      s = (s >= 0.f) ? s : 0.2f * s;
      m = fmaxf(m, s);
    }
  }
  m = warp_max(m);
  if (lane == 0) mout[i] = m;
}

// P[i,j] = mask ? exp(s - m_i) : 0 for j<n, 0 in pad cols; l[i] = row sum.
// P stride = KP (multiple of 4).  exp(-1e9 - m) == 0 in fp32, so this equals
// the reference softmax numerators exactly.
__global__ void attn_pexp_kernel(const float* __restrict__ Amat,
                                 const float* __restrict__ av,
                                 const float* __restrict__ bv,
                                 const float* __restrict__ mv,
                                 float* __restrict__ P,
                                 float* __restrict__ lout, int n, int KP) {
  const int i = (blockIdx.x * blockDim.x + threadIdx.x) >> 5;
  const int lane = threadIdx.x & 31;
  if (i >= n) return;
  const float* Ar = Amat + (size_t)i * n;
  float* Pr = P + (size_t)i * KP;
  const float ai = av[i], mi = mv[i];
  float sum = 0.f;
  for (int j = lane; j < KP; j += 32) {
    float p = 0.f;
    if (j < n && ((Ar[j] > 0.f) || (j == i))) {
      float s = ai + bv[j];
      s = (s >= 0.f) ? s : 0.2f * s;
      p = expf(s - mi);
    }
    Pr[j] = p;
    sum += p;
  }
  sum = warp_sum(sum);
  if (lane == 0) lout[i] = sum;
}

// score[i] = sigmoid((X[i].pw + pb)/100)
__global__ void pool_score_kernel(const float* __restrict__ X,
                                  const float* __restrict__ pw,
                                  const float* __restrict__ pb,
                                  float* __restrict__ score, int n) {
  const int i = (blockIdx.x * blockDim.x + threadIdx.x) >> 5;
  const int lane = threadIdx.x & 31;
  if (i >= n) return;
  const float* row = X + (size_t)i * FDIM;
  float s = 0.f;
  for (int c = lane; c < FDIM; c += 32) s += row[c] * pw[c];
  s = warp_sum(s);
  if (lane == 0) {
    const float t = (s + pb[0]) * 0.01f;
    score[i] = 1.f / (1.f + expf(-t));
  }
}

// selected[i] = 1 iff rank(i) < kn, rank = #{j : s_j > s_i or (== and j<i)}
__global__ void topk_rank_kernel(const float* __restrict__ score,
                                 int* __restrict__ selected, int n, int kn) {
  const int i = (blockIdx.x * blockDim.x + threadIdx.x) >> 5;
  const int lane = threadIdx.x & 31;
  if (i >= n) return;
  const float si = score[i];
  int c = 0;
  for (int j = lane; j < n; j += 32) {
    const float sj = score[j];
    if (sj > si || (sj == si && j < i)) ++c;
  }
  c = warp_sum_i(c);
  if (lane == 0) selected[i] = (c < kn) ? 1 : 0;
}

// compact selected indices in ascending order (GAT is permutation-equivariant
// so any consistent ordering of the selected set yields the same output)
__global__ void topk_compact_kernel(const float* __restrict__ score,
                                    const int* __restrict__ selected,
                                    int* __restrict__ idx,
                                    float* __restrict__ vals, int n) {
  const int i = (blockIdx.x * blockDim.x + threadIdx.x) >> 5;
  const int lane = threadIdx.x & 31;
  if (i >= n) return;
  if (selected[i] == 0) return;   // wave-uniform
  int c = 0;
  for (int j = lane; j < i; j += 32) c += selected[j];
  c = warp_sum_i(c);
  if (lane == 0) { idx[c] = i; vals[c] = score[i]; }
}

__global__ void gather_rows_kernel(const float* __restrict__ X,
                                   const int* __restrict__ idx,
                                   const float* __restrict__ vals,
                                   float* __restrict__ out, int kn, int f) {
  const int t = blockIdx.x * blockDim.x + threadIdx.x;
  if (t >= kn * f) return;
  const int r = t / f, c = t % f;
  out[t] = X[(size_t)idx[r] * f + c] * vals[r];
}

__global__ void gather_adj_kernel(const float* __restrict__ Abig, int nbig,
                                  const int* __restrict__ idx,
                                  float* __restrict__ Asm, int ns) {
  const long long t = (long long)blockIdx.x * blockDim.x + threadIdx.x;
  if (t >= (long long)ns * ns) return;
  const int r = (int)(t / ns), s = (int)(t % ns);
  Asm[t] = Abig[(size_t)idx[r] * nbig + idx[s]];
}

__global__ void zero_kernel(float* __restrict__ p, long long n) {
  const long long t = (long long)blockIdx.x * blockDim.x + threadIdx.x;
  if (t < n) p[t] = 0.f;
}

__global__ void scatter_rows_kernel(const float* __restrict__ Xs,
                                    const int* __restrict__ idx,
                                    float* __restrict__ out, int kn, int f) {
  const int t = blockIdx.x * blockDim.x + threadIdx.x;
  if (t >= kn * f) return;
  const int r = t / f, c = t % f;
  out[(size_t)idx[r] * f + c] = Xs[t];
}

__global__ void concat_kernel(const float* __restrict__ X1,
                              const float* __restrict__ X2,
                              float* __restrict__ out, int n, int f) {
  const int t = blockIdx.x * blockDim.x + threadIdx.x;
  if (t >= n * 2 * f) return;
  const int i = t / (2 * f), c = t % (2 * f);
  out[t] = (c < f) ? X1[(size_t)i * f + c] : X2[(size_t)i * f + (c - f)];
}

__global__ void copy_cols_kernel(const float* __restrict__ src,
                                 float* __restrict__ dst,
                                 int n, int fsrc, int fdst) {
  const int t = blockIdx.x * blockDim.x + threadIdx.x;
  if (t >= n * fdst) return;
  const int i = t / fdst, c = t % fdst;
  dst[t] = src[(size_t)i * fsrc + c];
}

// ---------------------------------------------------------------------------

static inline dim3 waves_grid(int waves) { return dim3((waves * 32 + 255) / 256); }
static inline dim3 elems_grid(long long n) { return dim3((unsigned)((n + 255) / 256)); }
static inline int align16(int n) { return (n + 15) & ~15; }
static inline int align4(int n)  { return (n + 3) & ~3; }

extern "C" void kernel_launch(void* const* d_in, const int* in_sizes, int n_in,
                              void* d_out, int out_size, void* d_ws, size_t ws_size,
                              hipStream_t stream) {
  const float* A0        = (const float*)d_in[0];
  const float* X         = (const float*)d_in[1];
  const float* start_W   = (const float*)d_in[2];
  const float* start_b   = (const float*)d_in[3];
  const float* start_phi = (const float*)d_in[4];
  const float* down_W    = (const float*)d_in[5];
  const float* down_b    = (const float*)d_in[6];
  const float* down_phi  = (const float*)d_in[7];
  const float* pool_w    = (const float*)d_in[8];
  const float* pool_b    = (const float*)d_in[9];
  const float* bottom_W  = (const float*)d_in[10];
  const float* bottom_b  = (const float*)d_in[11];
  const float* bottom_phi= (const float*)d_in[12];
  const float* up_W      = (const float*)d_in[13];
  const float* up_b      = (const float*)d_in[14];
  const float* up_phi    = (const float*)d_in[15];
  const float* end_W     = (const float*)d_in[16];
  const float* end_b     = (const float*)d_in[17];
  const float* end_phi   = (const float*)d_in[18];
  float* out = (float*)d_out;

  // ---- workspace bump allocator (row-padded feature buffers) ----
  size_t off = 0;
  char* base = (char*)d_ws;
  auto alloc_f = [&](size_t nelem) -> float* {
    off = (off + 255) & ~(size_t)255;
    float* p = (float*)(base + off);
    off += nelem * sizeof(float);
    return p;
  };
  auto alloc_i = [&](size_t nelem) -> int* {
    off = (off + 255) & ~(size_t)255;
    int* p = (int*)(base + off);
    off += nelem * sizeof(int);
    return p;
  };
  const int N1p = align16(N1), N2p = align16(N2);

  float* Xs    = alloc_f((size_t)N0 * FDIM);      // start_out / org_X
  float* Xd0   = alloc_f((size_t)N0 * FDIM);      // down_outs[0]
  float* X1    = alloc_f((size_t)N1p * FDIM);
  float* Xd1   = alloc_f((size_t)N1p * FDIM);     // down_outs[1]
  float* X2    = alloc_f((size_t)N2p * FDIM);
  float* Xb    = alloc_f((size_t)N2p * FDIM);
  float* Xu1   = alloc_f((size_t)N1p * FDIM);
  float* Xup0  = alloc_f((size_t)N1p * FDIM);
  float* Xu0   = alloc_f((size_t)N0 * FDIM);
  float* Xup1  = alloc_f((size_t)N0 * FDIM);
  float* Xcat  = alloc_f((size_t)N0 * 2 * FDIM);
  float* xp    = alloc_f((size_t)N0 * FDIM);      // padded scratch, reused
  float* avec  = alloc_f(N0);
  float* bvec  = alloc_f(N0);
  float* mvec  = alloc_f(N0);
  float* lvec  = alloc_f(N0);
  float* score = alloc_f(N0);
  float* vals0 = alloc_f(N1);
  float* vals1 = alloc_f(N2);
  float* A1    = alloc_f((size_t)N1 * N1);
  float* A2    = alloc_f((size_t)N2 * N2);
  float* P     = alloc_f((size_t)N0 * N0);        // strided KP per level
  int*   sel   = alloc_i(N0);
  int*   idx0  = alloc_i(N1);
  int*   idx1  = alloc_i(N2);
  (void)ws_size; (void)in_sizes; (void)n_in; (void)out_size;

  // ---- one GAT layer ----
  auto run_gat = [&](const float* Amat, int n, const float* Xin, int fin,
                     const float* W, const float* b, const float* phi,
                     const float* res, const float* extra, float* outX) {
    const int Mpad = align16(n);
    const int KP   = align4(n);
    if (Mpad > n) {  // zero xp pad rows so P_pad(0) * xp_pad(0) == 0
      zero_kernel<<<elems_grid((long long)(Mpad - n) * FDIM), 256, 0, stream>>>(
          xp + (size_t)n * FDIM, (long long)(Mpad - n) * FDIM);
    }
    const int stripes = FDIM / 32;                 // 10
    const int waves = (Mpad / 16) * stripes;
    gemm_bias_kernel<<<waves_grid(waves), 256, 0, stream>>>(
        Xin, W, b, xp, n, fin, FDIM);
    rowdot2_kernel<<<waves_grid(n), 256, 0, stream>>>(xp, phi, avec, bvec, n);
    attn_rowmax_kernel<<<waves_grid(n), 256, 0, stream>>>(Amat, avec, bvec, mvec, n);
    attn_pexp_kernel<<<waves_grid(n), 256, 0, stream>>>(Amat, avec, bvec, mvec,
                                                        P, lvec, n, KP);
    gemm_attn_kernel<<<waves_grid(waves), 256, 0, stream>>>(
        P, KP, xp, lvec, res, extra, outX, n, KP, FDIM);
  };

  // ---- pool level ----
  auto run_pool = [&](const float* Xlvl, const float* Alvl, int n, int kn,
                      const float* pw, const float* pb,
                      int* idx, float* vals, float* Xsm, float* Asm) {
    pool_score_kernel<<<waves_grid(n), 256, 0, stream>>>(Xlvl, pw, pb, score, n);
    topk_rank_kernel<<<waves_grid(n), 256, 0, stream>>>(score, sel, n, kn);
    topk_compact_kernel<<<waves_grid(n), 256, 0, stream>>>(score, sel, idx, vals, n);
    gather_rows_kernel<<<elems_grid((long long)kn * FDIM), 256, 0, stream>>>(
        Xlvl, idx, vals, Xsm, kn, FDIM);
    gather_adj_kernel<<<elems_grid((long long)kn * kn), 256, 0, stream>>>(
        Alvl, n, idx, Asm, kn);
  };

  // ======== pipeline ========
  // start (residual=false)
  run_gat(A0, N0, X, FDIM, start_W, start_b, start_phi, nullptr, nullptr, Xs);

  // down0 (residual=true) + pool0
  run_gat(A0, N0, Xs, FDIM, down_W, down_b, down_phi, Xs, nullptr, Xd0);
  run_pool(Xd0, A0, N0, N1, pool_w, pool_b, idx0, vals0, X1, A1);

  // down1 + pool1
  run_gat(A1, N1, X1, FDIM, down_W + FDIM * FDIM, down_b + FDIM,
          down_phi + 2 * FDIM, X1, nullptr, Xd1);
  run_pool(Xd1, A1, N1, N2, pool_w + FDIM, pool_b + 1, idx1, vals1, X2, A2);

  // bottom
  run_gat(A2, N2, X2, FDIM, bottom_W, bottom_b, bottom_phi, X2, nullptr, Xb);

  // up0: unpool to N1, GAT, + down_outs[1]
  zero_kernel<<<elems_grid((long long)N1p * FDIM), 256, 0, stream>>>(
      Xu1, (long long)N1p * FDIM);
  scatter_rows_kernel<<<elems_grid((long long)N2 * FDIM), 256, 0, stream>>>(
      Xb, idx1, Xu1, N2, FDIM);
  run_gat(A1, N1, Xu1, FDIM, up_W, up_b, up_phi, Xu1, Xd1, Xup0);

  // up1: unpool to N0, GAT, + down_outs[0]
  zero_kernel<<<elems_grid((long long)N0 * FDIM), 256, 0, stream>>>(
      Xu0, (long long)N0 * FDIM);
  scatter_rows_kernel<<<elems_grid((long long)N1 * FDIM), 256, 0, stream>>>(
      Xup0, idx0, Xu0, N1, FDIM);
  run_gat(A0, N0, Xu0, FDIM, up_W + FDIM * FDIM, up_b + FDIM,
          up_phi + 2 * FDIM, Xu0, Xd0, Xup1);

  // end: concat(X, org_X) -> GAT(f_in=640), write straight into d_out
  concat_kernel<<<elems_grid((long long)N0 * 2 * FDIM), 256, 0, stream>>>(
      Xup1, Xs, Xcat, N0, FDIM);
  run_gat(A0, N0, Xcat, 2 * FDIM, end_W, end_b, end_phi, nullptr, nullptr, out);

  // second output: start_out[:, :268]
  copy_cols_kernel<<<elems_grid((long long)N0 * 268), 256, 0, stream>>>(
      Xs, out + (size_t)N0 * FDIM, N0, FDIM, 268);
}